// Net_57312043598543
// MI455X (gfx1250) — compile-verified
//
#include <hip/hip_runtime.h>
#include <math.h>

typedef __attribute__((ext_vector_type(2))) float v2f;
typedef __attribute__((ext_vector_type(8))) float v8f;

#define TPB 256
#define F_IN 512
#define H1 16
#define NCLS 8

// ---------------------------------------------------------------------------
// Degree / normalization
// ---------------------------------------------------------------------------
__global__ void k_deg_init(float* __restrict__ deg, int n) {
    int i = blockIdx.x * blockDim.x + threadIdx.x;
    if (i < n) deg[i] = 1.0f;  // self-loop weight
}

__global__ void k_deg_scatter(const int* __restrict__ ei, const float* __restrict__ w,
                              float* __restrict__ deg, int E) {
    int e = blockIdx.x * blockDim.x + threadIdx.x;
    if (e < E) atomicAdd(&deg[ei[E + e]], w[e]);  // dst row of edge_index
}

__global__ void k_rsqrt_inplace(float* __restrict__ deg, int n) {
    int i = blockIdx.x * blockDim.x + threadIdx.x;
    if (i < n) {
        float d = deg[i];
        deg[i] = (d > 0.0f) ? rsqrtf(d) : 0.0f;
    }
}

__global__ void k_edge_norm(const int* __restrict__ ei, const float* __restrict__ w,
                            const float* __restrict__ dinv, float* __restrict__ norm, int E) {
    int e = blockIdx.x * blockDim.x + threadIdx.x;
    if (e < E) norm[e] = dinv[ei[e]] * w[e] * dinv[ei[E + e]];
}

// ---------------------------------------------------------------------------
// GEMM1: h1[n,16] = x[n,512] @ W1[512,16]   (exact fp32 via V_WMMA_F32_16X16X4_F32)
// One wave32 per 16-row tile; 8 waves (8 tiles = 128 rows) per block.
// W1 staged in LDS (32 KB), A streamed as float2 per lane per K-step.
// ---------------------------------------------------------------------------
__global__ __launch_bounds__(TPB) void k_gemm1_wmma(const float* __restrict__ x,
                                                    const float* __restrict__ W1,
                                                    float* __restrict__ h1, int n) {
    __shared__ float ldsW[F_IN * H1];  // 32 KB, row-major [k][n]

    // cooperative load of W1 into LDS (2048 float4 / 256 threads = 8 each)
    {
        const float4* w4 = reinterpret_cast<const float4*>(W1);
        float4* l4 = reinterpret_cast<float4*>(ldsW);
        for (int i = threadIdx.x; i < (F_IN * H1) / 4; i += TPB) l4[i] = w4[i];
    }
    __syncthreads();

    const int lane = threadIdx.x & 31;
    const int wave = threadIdx.x >> 5;
    const int tile = blockIdx.x * (TPB / 32) + wave;  // 16-row tile index
    const int tiles = (n + 15) >> 4;
    if (tile >= tiles) return;  // uniform per wave: EXEC stays all-ones inside

    const int row0 = tile * 16;
    const int m = lane & 15;            // matrix row (A) / column (B,C)
    const int khalf = (lane >> 4) * 2;  // lanes 0-15 -> K=0,1 ; lanes 16-31 -> K=2,3

    // clamp row for tail-tile loads (stores are guarded)
    int arow = row0 + m;
    if (arow >= n) arow = n - 1;
    const float* aptr = x + (size_t)arow * F_IN + khalf;

    v8f c = {};
#pragma unroll 4
    for (int k0 = 0; k0 < F_IN; k0 += 4) {
        // A fragment: ISA 16x4 f32 layout
        const float2 av = *reinterpret_cast<const float2*>(aptr + k0);
        v2f a; a.x = av.x; a.y = av.y;
        // B fragment: row k striped across 16 lanes within a VGPR
        v2f b;
        b.x = ldsW[(k0 + khalf + 0) * H1 + m];
        b.y = ldsW[(k0 + khalf + 1) * H1 + m];
        c = __builtin_amdgcn_wmma_f32_16x16x4_f32(
                /*neg_a=*/false, a, /*neg_b=*/false, b,
                /*c_mod=*/(short)0, c, /*reuse_a=*/false, /*reuse_b=*/false);
    }

    // C layout: VGPR r -> M = r (+8 for lanes 16-31), N = lane&15
    const int mbase = (lane < 16) ? 0 : 8;
#pragma unroll
    for (int r = 0; r < 8; ++r) {
        int row = row0 + mbase + r;
        if (row < n) h1[(size_t)row * H1 + m] = c[r];
    }
}

// ---------------------------------------------------------------------------
// Layer-1 aggregation:  hagg[dst] = dinv[dst]^2 * h1[dst]  +  sum_e norm*h1[src]
// ---------------------------------------------------------------------------
__global__ void k_agg1_init(const float* __restrict__ h1, const float* __restrict__ dinv,
                            float* __restrict__ hagg, int n) {
    int i = blockIdx.x * blockDim.x + threadIdx.x;  // over n*4 float4 quads
    if (i >= n * 4) return;
    int node = i >> 2;
    float d = dinv[node];
    float s = d * d;  // self-loop norm
    float4 v = reinterpret_cast<const float4*>(h1)[i];
    float4 o; o.x = s * v.x; o.y = s * v.y; o.z = s * v.z; o.w = s * v.w;
    reinterpret_cast<float4*>(hagg)[i] = o;
}

__global__ void k_agg1_edges(const int* __restrict__ ei, const float* __restrict__ norm,
                             const float* __restrict__ h1, float* __restrict__ hagg, int E) {
    int tid = blockIdx.x * blockDim.x + threadIdx.x;  // E*4 threads, float4 each
    if (tid >= E * 4) return;
    int e = tid >> 2, q = tid & 3;
    int src = ei[e], dst = ei[E + e];
    float nm = norm[e];
    float4 v = *reinterpret_cast<const float4*>(h1 + (size_t)src * H1 + q * 4);
    float* o = hagg + (size_t)dst * H1 + q * 4;
    atomicAdd(o + 0, nm * v.x);
    atomicAdd(o + 1, nm * v.y);
    atomicAdd(o + 2, nm * v.z);
    atomicAdd(o + 3, nm * v.w);
}

// ---------------------------------------------------------------------------
// Layer 2 linear: y[n,8] = relu(hagg + b1) @ W2[16,8]   (tiny K -> plain VALU)
// ---------------------------------------------------------------------------
__global__ __launch_bounds__(TPB) void k_layer2(const float* __restrict__ hagg,
                                                const float* __restrict__ b1,
                                                const float* __restrict__ W2,
                                                float* __restrict__ y, int n) {
    __shared__ float w2s[H1 * NCLS];
    __shared__ float b1s[H1];
    if (threadIdx.x < H1 * NCLS) w2s[threadIdx.x] = W2[threadIdx.x];
    if (threadIdx.x < H1) b1s[threadIdx.x] = b1[threadIdx.x];
    __syncthreads();

    int i = blockIdx.x * blockDim.x + threadIdx.x;
    if (i >= n) return;

    float z[H1];
    const float4* hr = reinterpret_cast<const float4*>(hagg + (size_t)i * H1);
#pragma unroll
    for (int q = 0; q < 4; ++q) {
        float4 v = hr[q];
        z[q * 4 + 0] = fmaxf(v.x + b1s[q * 4 + 0], 0.0f);
        z[q * 4 + 1] = fmaxf(v.y + b1s[q * 4 + 1], 0.0f);
        z[q * 4 + 2] = fmaxf(v.z + b1s[q * 4 + 2], 0.0f);
        z[q * 4 + 3] = fmaxf(v.w + b1s[q * 4 + 3], 0.0f);
    }
    float acc[NCLS];
#pragma unroll
    for (int c = 0; c < NCLS; ++c) acc[c] = 0.0f;
#pragma unroll
    for (int j = 0; j < H1; ++j)
#pragma unroll
        for (int c = 0; c < NCLS; ++c) acc[c] = fmaf(z[j], w2s[j * NCLS + c], acc[c]);

    float4* yo = reinterpret_cast<float4*>(y + (size_t)i * NCLS);
    float4 o0, o1;
    o0.x = acc[0]; o0.y = acc[1]; o0.z = acc[2]; o0.w = acc[3];
    o1.x = acc[4]; o1.y = acc[5]; o1.z = acc[6]; o1.w = acc[7];
    yo[0] = o0; yo[1] = o1;
}

// ---------------------------------------------------------------------------
// Layer-2 aggregation
// ---------------------------------------------------------------------------
__global__ void k_agg2_init(const float* __restrict__ y, const float* __restrict__ dinv,
                            float* __restrict__ agg2, int n) {
    int i = blockIdx.x * blockDim.x + threadIdx.x;  // n*2 quads
    if (i >= n * 2) return;
    int node = i >> 1;
    float d = dinv[node];
    float s = d * d;
    float4 v = reinterpret_cast<const float4*>(y)[i];
    float4 o; o.x = s * v.x; o.y = s * v.y; o.z = s * v.z; o.w = s * v.w;
    reinterpret_cast<float4*>(agg2)[i] = o;
}

__global__ void k_agg2_edges(const int* __restrict__ ei, const float* __restrict__ norm,
                             const float* __restrict__ y, float* __restrict__ agg2, int E) {
    int tid = blockIdx.x * blockDim.x + threadIdx.x;  // E*2 threads
    if (tid >= E * 2) return;
    int e = tid >> 1, q = tid & 1;
    int src = ei[e], dst = ei[E + e];
    float nm = norm[e];
    float4 v = *reinterpret_cast<const float4*>(y + (size_t)src * NCLS + q * 4);
    float* o = agg2 + (size_t)dst * NCLS + q * 4;
    atomicAdd(o + 0, nm * v.x);
    atomicAdd(o + 1, nm * v.y);
    atomicAdd(o + 2, nm * v.z);
    atomicAdd(o + 3, nm * v.w);
}

// ---------------------------------------------------------------------------
// Bias + log_softmax over 8 classes
// ---------------------------------------------------------------------------
__global__ void k_logsoftmax(const float* __restrict__ agg2, const float* __restrict__ b2,
                             float* __restrict__ out, int n) {
    int i = blockIdx.x * blockDim.x + threadIdx.x;
    if (i >= n) return;
    float z[NCLS];
    const float4* r = reinterpret_cast<const float4*>(agg2 + (size_t)i * NCLS);
    float4 v0 = r[0], v1 = r[1];
    z[0] = v0.x + b2[0]; z[1] = v0.y + b2[1]; z[2] = v0.z + b2[2]; z[3] = v0.w + b2[3];
    z[4] = v1.x + b2[4]; z[5] = v1.y + b2[5]; z[6] = v1.z + b2[6]; z[7] = v1.w + b2[7];
    float m = z[0];
#pragma unroll
    for (int c = 1; c < NCLS; ++c) m = fmaxf(m, z[c]);
    float s = 0.0f;
#pragma unroll
    for (int c = 0; c < NCLS; ++c) s += expf(z[c] - m);
    float l = m + logf(s);
    float4* o = reinterpret_cast<float4*>(out + (size_t)i * NCLS);
    float4 o0, o1;
    o0.x = z[0] - l; o0.y = z[1] - l; o0.z = z[2] - l; o0.w = z[3] - l;
    o1.x = z[4] - l; o1.y = z[5] - l; o1.z = z[6] - l; o1.w = z[7] - l;
    o[0] = o0; o[1] = o1;
}

// ---------------------------------------------------------------------------
// Host launcher
// ---------------------------------------------------------------------------
extern "C" void kernel_launch(void* const* d_in, const int* in_sizes, int n_in,
                              void* d_out, int out_size, void* d_ws, size_t ws_size,
                              hipStream_t stream) {
    const float* x  = (const float*)d_in[0];
    const int*   ei = (const int*)d_in[1];   // edge_index [2,E] flat
    const float* w  = (const float*)d_in[2];
    const float* W1 = (const float*)d_in[3];
    const float* b1 = (const float*)d_in[4];
    const float* W2 = (const float*)d_in[5];
    const float* b2 = (const float*)d_in[6];
    float* out = (float*)d_out;

    const int n = in_sizes[0] / F_IN;
    const int E = in_sizes[1] / 2;

    // workspace layout (256B aligned)
    char* p = (char*)d_ws;
    auto take = [&](size_t bytes) {
        char* q = p;
        p += (bytes + 255) & ~(size_t)255;
        return q;
    };
    float* dinv = (float*)take((size_t)n * 4);            // deg -> dinv in place
    float* norm = (float*)take((size_t)E * 4);
    float* h1   = (float*)take((size_t)n * H1 * 4);
    float* hagg = (float*)take((size_t)n * H1 * 4);
    float* y    = (float*)take((size_t)n * NCLS * 4);
    float* agg2 = (float*)take((size_t)n * NCLS * 4);
    (void)ws_size; (void)n_in; (void)out_size;

    auto nb = [](long long t) { return (int)((t + TPB - 1) / TPB); };

    k_deg_init     <<<nb(n), TPB, 0, stream>>>(dinv, n);
    k_deg_scatter  <<<nb(E), TPB, 0, stream>>>(ei, w, dinv, E);
    k_rsqrt_inplace<<<nb(n), TPB, 0, stream>>>(dinv, n);
    k_edge_norm    <<<nb(E), TPB, 0, stream>>>(ei, w, dinv, norm, E);

    const int tiles = (n + 15) / 16;
    k_gemm1_wmma   <<<(tiles + (TPB / 32) - 1) / (TPB / 32), TPB, 0, stream>>>(x, W1, h1, n);

    k_agg1_init    <<<nb((long long)n * 4), TPB, 0, stream>>>(h1, dinv, hagg, n);
    k_agg1_edges   <<<nb((long long)E * 4), TPB, 0, stream>>>(ei, norm, h1, hagg, E);

    k_layer2       <<<nb(n), TPB, 0, stream>>>(hagg, b1, W2, y, n);

    k_agg2_init    <<<nb((long long)n * 2), TPB, 0, stream>>>(y, dinv, agg2, n);
    k_agg2_edges   <<<nb((long long)E * 2), TPB, 0, stream>>>(ei, norm, y, agg2, E);

    k_logsoftmax   <<<nb(n), TPB, 0, stream>>>(agg2, b2, out, n);
}